// GPUAugmentation_24120536335026
// MI455X (gfx1250) — compile-verified
//
#include <hip/hip_runtime.h>
#include <stdint.h>

// ---------------------------------------------------------------------------
// GPU augmentation: flip / grayscale / brightness / contrast / clip
// x: (256, 3, 224, 224) f32.  Pure bandwidth problem (~1.3 flop/byte):
//   pass 1: per-(b,c) raw means (also warms 192MB L2 with the 154MB tensor)
//   pass 2: elementwise transform, staged through LDS via
//           global_load_async_to_lds_b128 (ASYNCcnt pipeline), NT stores.
// Transformed mean derived analytically from raw means:
//   mean_y[b,c] = br[b] * (gray[b] ? dot(w, m_raw[b,:]) : m_raw[b,c])
// ---------------------------------------------------------------------------

typedef float v4f __attribute__((ext_vector_type(4)));

#define Bn    256
#define Cn    3
#define Hn    224
#define Wn    224
#define ROW4  (Wn / 4)          // 56 float4 per row
#define HW4   (Hn * ROW4)       // 12544 float4 per channel
#define SPLIT 7                 // blocks per batch element in pass 2
#define TILES 7                 // tiles per block (TILES*256 = HW4/SPLIT)
#define CHUNK (HW4 / SPLIT)     // 1792 float4 positions per block per channel

// ---------------------------- pass 1: raw means ----------------------------
__global__ __launch_bounds__(256)
void mean_kernel(const float* __restrict__ x, float* __restrict__ means) {
  const int bc  = blockIdx.x;          // 0 .. B*C-1
  const int tid = threadIdx.x;
  const v4f* xv = (const v4f*)x;
  const size_t base = (size_t)bc * HW4;

  float s = 0.0f;
  #pragma unroll 7
  for (int k = 0; k < 49; ++k) {       // 49*256 = 12544
    v4f v = xv[base + (size_t)(tid + k * 256)];
    s += (v.x + v.y) + (v.z + v.w);
  }

  __shared__ float red[256];
  red[tid] = s;
  __syncthreads();
  #pragma unroll
  for (int off = 128; off > 0; off >>= 1) {
    if (tid < off) red[tid] += red[tid + off];
    __syncthreads();
  }
  if (tid == 0) means[bc] = red[0] * (1.0f / (float)(Hn * Wn));
}

// ------------------------- pass 2: fused transform -------------------------
__global__ __launch_bounds__(256)
void aug_kernel(const float* __restrict__ x,
                const float* __restrict__ brightness,
                const float* __restrict__ contrastf,
                const int*   __restrict__ flip_mask,
                const int*   __restrict__ gray_mask,
                const int*   __restrict__ contrast_apply,
                const float* __restrict__ means,
                float* __restrict__ out) {
  __shared__ v4f stage[2][3][256];     // double-buffered, 24 KB

  const int b     = blockIdx.x;
  const int slice = blockIdx.y;
  const int tid   = threadIdx.x;

  const bool flip = flip_mask[b]      != 0;
  const bool gray = gray_mask[b]      != 0;
  const bool capp = contrast_apply[b] != 0;
  const float br  = brightness[b];
  const float cf  = contrastf[b];

  // transformed per-channel means (block-uniform scalars)
  float m0 = means[b * 3 + 0];
  float m1 = means[b * 3 + 1];
  float m2 = means[b * 3 + 2];
  if (gray) {
    float g = 0.2989f * m0 + 0.587f * m1 + 0.114f * m2;
    m0 = m1 = m2 = g;
  }
  m0 *= br; m1 *= br; m2 *= br;

  const v4f* xv = (const v4f*)x;
  v4f*       ov = (v4f*)out;
  const size_t cb = (size_t)b * Cn * HW4;

  // issue one tile: 3 channels x 256 lanes of b128 async copies into LDS
  auto issue = [&](int t, int buf) {
    const int p  = slice * CHUNK + t * 256 + tid;
    const int h  = p / ROW4;
    const int q  = p - h * ROW4;
    const int qs = flip ? (ROW4 - 1 - q) : q;  // mirrored source float4
    #pragma unroll
    for (int c = 0; c < Cn; ++c) {
      const v4f* src = xv + (cb + (size_t)c * HW4 + (size_t)h * ROW4 + qs);
      uint32_t lds = (uint32_t)(uintptr_t)&stage[buf][c][tid];
      asm volatile("global_load_async_to_lds_b128 %0, %1, off"
                   :: "v"(lds), "v"(src) : "memory");
    }
  };

  issue(0, 0);
  for (int t = 0; t < TILES; ++t) {
    if (t + 1 < TILES) {
      issue(t + 1, (t + 1) & 1);
      // in-order completion: <=3 outstanding means tile t has landed
      asm volatile("s_wait_asynccnt 0x3" ::: "memory");
    } else {
      asm volatile("s_wait_asynccnt 0x0" ::: "memory");
    }

    const int buf = t & 1;
    v4f v0 = stage[buf][0][tid];
    v4f v1 = stage[buf][1][tid];
    v4f v2 = stage[buf][2][tid];

    if (flip) {  // mirrored float4 -> reverse components
      v0 = v4f{v0.w, v0.z, v0.y, v0.x};
      v1 = v4f{v1.w, v1.z, v1.y, v1.x};
      v2 = v4f{v2.w, v2.z, v2.y, v2.x};
    }
    if (gray) {
      v4f g = 0.2989f * v0 + 0.587f * v1 + 0.114f * v2;
      v0 = g; v1 = g; v2 = g;
    }
    v0 *= br; v1 *= br; v2 *= br;
    if (capp) {
      v0 = (v0 - m0) * cf + m0;
      v1 = (v1 - m1) * cf + m1;
      v2 = (v2 - m2) * cf + m2;
    }
    // clip to [-2.5, 2.5]
    #pragma unroll
    for (int i = 0; i < 4; ++i) {
      v0[i] = fminf(fmaxf(v0[i], -2.5f), 2.5f);
      v1[i] = fminf(fmaxf(v1[i], -2.5f), 2.5f);
      v2[i] = fminf(fmaxf(v2[i], -2.5f), 2.5f);
    }

    const int p = slice * CHUNK + t * 256 + tid;   // natural output position
    __builtin_nontemporal_store(v0, &ov[cb + 0 * (size_t)HW4 + p]);
    __builtin_nontemporal_store(v1, &ov[cb + 1 * (size_t)HW4 + p]);
    __builtin_nontemporal_store(v2, &ov[cb + 2 * (size_t)HW4 + p]);
  }
}

// ------------------------------- launcher ----------------------------------
extern "C" void kernel_launch(void* const* d_in, const int* in_sizes, int n_in,
                              void* d_out, int out_size, void* d_ws, size_t ws_size,
                              hipStream_t stream) {
  const float* x  = (const float*)d_in[0];
  const float* br = (const float*)d_in[1];
  const float* cf = (const float*)d_in[2];
  const int*   fm = (const int*)d_in[3];
  const int*   gm = (const int*)d_in[4];
  const int*   ca = (const int*)d_in[5];
  float* out   = (float*)d_out;
  float* means = (float*)d_ws;   // B*C = 768 floats of scratch

  mean_kernel<<<dim3(Bn * Cn), dim3(256), 0, stream>>>(x, means);
  aug_kernel<<<dim3(Bn, SPLIT), dim3(256), 0, stream>>>(x, br, cf, fm, gm, ca,
                                                        means, out);
}